// OneDimEquivalent_18296560681338
// MI455X (gfx1250) — compile-verified
//
#include <hip/hip_runtime.h>
#include <stdint.h>
#include <math.h>

// ---------------------------------------------------------------------------
// OneDimEquivalent scan for MI455X (gfx1250), single-workgroup design.
//
//  Phase A: TDM (tensor_load_to_lds) streams u[32768] (128KB) global -> LDS.
//  Phase B: 64 lanes compute Gauss-Legendre nodes/weights (Newton on P_64).
//  Phase C: 32 waves build a 4096-entry LUT of
//              g(delta) = (1/2pi) * sum_q ew[q] * sech^2(delta*x_q)
//           using chained V_WMMA_F32_16X16X4_F32 (16 deltas / wave / chain,
//           K=64 accumulated over 16 WMMAs, full f32 precision).
//           sech^2 is branchless: one v_exp_f32 + one v_rcp_f32, so EXEC
//           stays all-1s and no IEEE-division macro-ops pollute the chain.
//  Phase D: one lane runs the 32768-step recurrence out of LDS with a
//           linear-interp LUT lookup instead of 64 tanh per step.
// ---------------------------------------------------------------------------

typedef __attribute__((ext_vector_type(2))) float        v2f;
typedef __attribute__((ext_vector_type(8))) float        v8f;
typedef __attribute__((ext_vector_type(4))) unsigned int v4u;
typedef __attribute__((ext_vector_type(8))) int          v8i;
typedef __attribute__((ext_vector_type(4))) int          v4i;

#define SEQ_LEN     32768
#define QORD        64
#define LUT_N       4096
#define DMAX        32.0f
#define A_HALF      5.0f          // quadrature interval [-A, A]
#define SIG_M       1.0f
#define SIG_I       1.0f
#define SIG_MN      1.4f
#define SIG_NI      2.6f
#define DT_OVER_TAU 0.2f          // DT/TAU = 20/100
#define INV_2PI     0.15915494309189535f
#define PI_F        3.14159265358979323846f

// Branchless sech^2(x) = 1 - tanh^2(x) = 4 e^{-2|x|} / (1 + e^{-2|x|})^2.
// One v_exp_f32 + one v_rcp_f32; e -> 0 for large |x| so it is stable.
__device__ __forceinline__ float sech2f(float x) {
    float e = __expf(-2.0f * fabsf(x));
    float d = 1.0f + e;
    return 4.0f * e * __builtin_amdgcn_rcpf(d * d);
}

__global__ __launch_bounds__(1024)
void onedim_scan_kernel(const float* __restrict__ u, float* __restrict__ out) {
    extern __shared__ char smem[];
    float* lds_u = (float*)smem;          // SEQ_LEN floats   (128 KB)
    float* lut   = lds_u + SEQ_LEN;       // LUT_N floats     (16 KB)
    float* xs    = lut + LUT_N;           // QORD floats
    float* ew    = xs + QORD;             // QORD floats

    const int tid  = threadIdx.x;
    const int lane = tid & 31;
    const int wave = tid >> 5;

    // ---- Phase A: TDM DMA of u -> LDS, issued by the last wave ------------
    if (wave == 31) {
        unsigned long long ga = (unsigned long long)(uintptr_t)u;
        // Low 32 bits of a flat LDS-aperture address are the LDS byte offset.
        unsigned int lds_off = (unsigned int)(uintptr_t)(void*)lds_u;

        v4u g0;
        g0.x = 1u;                                   // count = 1 (valid D#)
        g0.y = lds_off;                              // lds_addr (bytes)
        g0.z = (unsigned int)(ga & 0xFFFFFFFFu);     // global_addr[31:0]
        g0.w = ((unsigned int)(ga >> 32) & 0x01FFFFFFu) | (2u << 30); // [56:32] | type=2

        v8i g1;
        g1[0] = (int)(2u << 16);      // data_size = 2 -> 4-byte elements
        g1[1] = (int)0x80000000u;     // tensor_dim0[15:0] = 32768 (bits 63:48 -> [31:16])
        g1[2] = (int)0x00010000u;     // tensor_dim0[31:16]=0, tensor_dim1 = 1
        g1[3] = (int)0x80000000u;     // tile_dim0 = 32768 (bits 127:112)
        g1[4] = 1;                    // tile_dim1 = 1, tile_dim2 = 0
        g1[5] = 32768;                // tensor_dim0_stride[31:0]
        g1[6] = 0;                    // stride hi / tensor_dim1_stride lo
        g1[7] = 0;

        v4i z4 = {0, 0, 0, 0};
        v8i z8 = {0, 0, 0, 0, 0, 0, 0, 0};
        __builtin_amdgcn_tensor_load_to_lds(g0, g1, z4, z4, z8, 0);
        __builtin_amdgcn_s_wait_tensorcnt(0);
    }

    // ---- Phase B: Gauss-Legendre nodes/weights (Newton, fixed count) ------
    if (tid < QORD) {
        float x = __cosf(PI_F * ((float)(tid + 1) - 0.25f) / ((float)QORD + 0.5f));
        float p0, p1, dp;
        for (int it = 0; it < 60; ++it) {
            p0 = 1.0f; p1 = x;
            for (int l = 2; l <= QORD; ++l) {
                float p2 = ((2.0f * (float)l - 1.0f) * x * p1 - ((float)l - 1.0f) * p0) / (float)l;
                p0 = p1; p1 = p2;
            }
            dp = (float)QORD * (x * p1 - p0) / (x * x - 1.0f);
            x  = x - p1 / dp;
        }
        // final derivative at converged x for the weight
        p0 = 1.0f; p1 = x;
        for (int l = 2; l <= QORD; ++l) {
            float p2 = ((2.0f * (float)l - 1.0f) * x * p1 - ((float)l - 1.0f) * p0) / (float)l;
            p0 = p1; p1 = p2;
        }
        dp = (float)QORD * (x * p1 - p0) / (x * x - 1.0f);
        float w = 2.0f / ((1.0f - x * x) * dp * dp);
        float z = A_HALF * x;                       // node scaled to [-A, A]
        xs[tid] = z;
        ew[tid] = A_HALF * w * __expf(-0.5f * z * z); // fold Gaussian + scale
    }
    __syncthreads();

    // ---- Phase C: LUT via chained V_WMMA_F32_16X16X4_F32 ------------------
    // A (16x4 f32): row-replicated quadrature weights ew[4j .. 4j+3]
    //   lane layout: lanes 0-15 VGPR{0,1} = K{0,1}; lanes 16-31 = K{2,3}
    // B (4x16 f32): sech^2(delta_n * x_k), col n = lane&15, same K split.
    // After 16 WMMAs (K = 64), every row of D holds the quadrature sum.
    {
        const int   col    = lane & 15;
        const int   hi     = lane >> 4;
        const float dscale = DMAX / (float)(LUT_N - 1);

        for (int grp = wave; grp < LUT_N / 16; grp += 32) {   // uniform per wave
            float delta = (float)(grp * 16 + col) * dscale;
            v8f acc = {0.f, 0.f, 0.f, 0.f, 0.f, 0.f, 0.f, 0.f};
#pragma unroll
            for (int j = 0; j < 16; ++j) {
                int q = 4 * j + 2 * hi;
                v2f a, b;
                a.x = ew[q];
                a.y = ew[q + 1];
                b.x = sech2f(delta * xs[q]);
                b.y = sech2f(delta * xs[q + 1]);
                acc = __builtin_amdgcn_wmma_f32_16x16x4_f32(
                          false, a, false, b, (short)0, acc, false, false);
            }
            if (hi == 0) lut[grp * 16 + col] = acc[0] * INV_2PI;  // D row 0
        }
    }
    __syncthreads();

    // ---- Phase D: serial 32768-step recurrence out of LDS -----------------
    if (tid == 0) {
        float k = 0.0f, v = 0.0f;
        const float idx_scale = (float)(LUT_N - 1) / DMAX;
        for (int t = 0; t < SEQ_LEN; ++t) {
            float ut    = lds_u[t];
            float delta = sqrtf(SIG_M * SIG_M * k * k + SIG_I * SIG_I * ut * ut);
            float ft    = fminf(delta * idx_scale, (float)(LUT_N - 2));
            int   i0    = (int)ft;
            float fr    = ft - (float)i0;
            float g0v   = lut[i0];
            float g     = g0v + fr * (lut[i0 + 1] - g0v);
            float dk = (-k + SIG_MN * g * k + SIG_NI * g * v) * DT_OVER_TAU;
            k += dk;
            v += (-v + ut) * DT_OVER_TAU;
        }
        out[0] = tanhf(k);
    }
}

extern "C" void kernel_launch(void* const* d_in, const int* in_sizes, int n_in,
                              void* d_out, int out_size, void* d_ws, size_t ws_size,
                              hipStream_t stream) {
    (void)in_sizes; (void)n_in; (void)out_size; (void)d_ws; (void)ws_size;
    const float* u  = (const float*)d_in[0];
    float*      out = (float*)d_out;
    size_t smem = (size_t)(SEQ_LEN + LUT_N + 2 * QORD) * sizeof(float); // ~145 KB < 320 KB/WGP
    onedim_scan_kernel<<<1, 1024, smem, stream>>>(u, out);
}